// GFCNG_83863531422351
// MI455X (gfx1250) — compile-verified
//
#include <hip/hip_runtime.h>
#include <math.h>

// ---------------------------------------------------------------------------
// CDNA5 (gfx1250) implementation of the GFCNG graph U-Net forward pass.
// Dense node transforms use V_WMMA_F32_16X16X4_F32 (fp32 matrix pipe,
// full precision). Edge message passing = L2-resident gather + atomic
// scatter (fits in the 192MB L2). TopK via O(n^2) rank counting.
// ---------------------------------------------------------------------------

typedef float v2f __attribute__((ext_vector_type(2)));
typedef float v8f __attribute__((ext_vector_type(8)));

#define N0   16384
#define EDG  262144

// ---------------------------- kernels --------------------------------------

__global__ void fill_f32(float* __restrict__ p, float v, int n) {
  int i = blockIdx.x * blockDim.x + threadIdx.x;
  if (i < n) p[i] = v;
}

// C[M,N] = A[M,K] * B[K,N]; row-major with leading dims. One wave per 16x16
// C tile, fp32 WMMA with K-step 4. A layout (ISA 7.12.2, 32-bit A 16x4):
// lanes 0-15 hold M=lane, VGPR0/1 = K,K+1; lanes 16-31 hold K+2,K+3.
// B (4x16) mirrored: lanes hold N=lane&15, VGPRs hold successive K rows.
// C: 8 VGPRs, lane<16 -> M=i, lane>=16 -> M=i+8, N=lane&15.
// Fast path (interior tile, K%4==0, even lda): unconditional loads, A as one
// 8-byte v2f load per step, unrolled 4x -> 4 WMMAs in flight, no exec churn.
__global__ __launch_bounds__(256) void gemm_wmma_f32(
    const float* __restrict__ A, int lda,
    const float* __restrict__ B, int ldb,
    float* __restrict__ C, int ldc,
    int M, int N, int K)
{
  int tilesN = (N + 15) >> 4;
  int tilesM = (M + 15) >> 4;
  int ntiles = tilesM * tilesN;
  int wave = blockIdx.x * (blockDim.x >> 5) + (threadIdx.x >> 5);
  if (wave >= ntiles) return;                 // wave-uniform exit
  int lane = threadIdx.x & 31;
  int half = lane >> 4;                       // 0: lanes 0-15, 1: lanes 16-31
  int l15  = lane & 15;
  int tm = wave / tilesN;
  int tn = wave - tm * tilesN;
  int row0 = tm << 4;
  int col0 = tn << 4;

  v8f acc = {};
  bool fast = (row0 + 16 <= M) && (col0 + 16 <= N) &&
              ((K & 3) == 0) && ((lda & 1) == 0);

  if (fast) {
    // lane-fixed base pointers; per-step offsets are uniform (kk)
    const float* Ap = A + (size_t)(row0 + l15) * lda + (half << 1);
    const float* Bp = B + (size_t)(half << 1) * ldb + (col0 + l15);
#pragma unroll 4
    for (int kk = 0; kk < K; kk += 4) {
      v2f a = *(const v2f*)(Ap + kk);         // A[M, ka..ka+1] contiguous
      v2f b;
      b.x = Bp[(size_t)kk * ldb];
      b.y = Bp[(size_t)(kk + 1) * ldb];
      acc = __builtin_amdgcn_wmma_f32_16x16x4_f32(
          false, a, false, b, (short)0, acc, false, false);
    }
    float* Cp = C + (size_t)(row0 + (half << 3)) * ldc + (col0 + l15);
#pragma unroll
    for (int i = 0; i < 8; ++i) Cp[(size_t)i * ldc] = acc[i];
  } else {
    int ar = row0 + l15;
    int bc = col0 + l15;
    for (int kk = 0; kk < K; kk += 4) {
      int ka = kk + (half << 1);              // this half covers ka, ka+1
      v2f a, b;
      a.x = (ar < M && ka     < K) ? A[(size_t)ar * lda + ka]     : 0.0f;
      a.y = (ar < M && ka + 1 < K) ? A[(size_t)ar * lda + ka + 1] : 0.0f;
      b.x = (bc < N && ka     < K) ? B[(size_t)(ka    ) * ldb + bc] : 0.0f;
      b.y = (bc < N && ka + 1 < K) ? B[(size_t)(ka + 1) * ldb + bc] : 0.0f;
      acc = __builtin_amdgcn_wmma_f32_16x16x4_f32(
          false, a, false, b, (short)0, acc, false, false);
    }
    if (bc < N) {
#pragma unroll
      for (int i = 0; i < 8; ++i) {
        int r = row0 + i + (half << 3);
        if (r < M) C[(size_t)r * ldc + bc] = acc[i];
      }
    }
  }
}

// Per (edge, channel): bilinear spline taps on xw[src], atomic scatter to dst.
__global__ void edge_msg(const float* __restrict__ xw,
                         const int* __restrict__ ei,
                         const float* __restrict__ ea,
                         float* __restrict__ agg, float* __restrict__ deg,
                         int E, int cout, int K, int ktot)
{
  long long tid = (long long)blockIdx.x * blockDim.x + threadIdx.x;
  long long total = (long long)E * cout;
  if (tid >= total) return;
  int e = (int)(tid / cout);
  int c = (int)(tid - (long long)e * cout);
  int src = ei[e];
  int dst = ei[E + e];
  const float* xr = xw + (size_t)src * ktot * cout;
  float val;
  if (K == 1) {
    val = xr[c];
  } else {
    float v0 = ea[2 * e]     * (float)(K - 1);
    float v1 = ea[2 * e + 1] * (float)(K - 1);
    float f0 = floorf(v0), f1 = floorf(v1);
    float r0 = v0 - f0,     r1 = v1 - f1;
    int   i0 = (int)f0,     i1 = (int)f1;
    val = 0.0f;
#pragma unroll
    for (int o1 = 0; o1 < 2; ++o1)
#pragma unroll
      for (int o0 = 0; o0 < 2; ++o0) {
        int j0 = i0 + o0; j0 = j0 < 0 ? 0 : (j0 > K - 1 ? K - 1 : j0);
        int j1 = i1 + o1; j1 = j1 < 0 ? 0 : (j1 > K - 1 ? K - 1 : j1);
        float wgt = (o0 ? r0 : 1.0f - r0) * (o1 ? r1 : 1.0f - r1);
        val += wgt * xr[(size_t)(j0 + K * j1) * cout + c];
      }
  }
  atomicAdd(&agg[(size_t)dst * cout + c], val);
  if (c == 0) atomicAdd(&deg[dst], 1.0f);
}

// out = [add?] elu?( agg/max(deg,1) + root + bias )
__global__ void finalize_conv(float* __restrict__ out,
                              const float* __restrict__ agg,
                              const float* __restrict__ deg,
                              const float* __restrict__ rootb,
                              const float* __restrict__ bias,
                              int n, int cout, int do_elu, int do_add)
{
  int tid = blockIdx.x * blockDim.x + threadIdx.x;
  if (tid >= n * cout) return;
  int i = tid / cout;
  int c = tid - i * cout;
  float d = fmaxf(deg[i], 1.0f);
  float v = agg[tid] / d + rootb[tid] + bias[c];
  if (do_elu) v = (v > 0.0f) ? v : (__expf(v) - 1.0f);
  out[tid] = do_add ? (out[tid] + v) : v;
}

__global__ void bn_apply(float* __restrict__ x,
                         const float* __restrict__ g, const float* __restrict__ b,
                         const float* __restrict__ m, const float* __restrict__ v,
                         int n, int cout)
{
  int tid = blockIdx.x * blockDim.x + threadIdx.x;
  if (tid >= n * cout) return;
  int c = tid % cout;
  x[tid] = g[c] * (x[tid] - m[c]) * rsqrtf(v[c] + 1e-5f) + b[c];
}

__global__ void wnorm_k(const float* __restrict__ w, int c, float* __restrict__ out) {
  __shared__ float sh[256];
  float s = 0.0f;
  for (int i = threadIdx.x; i < c; i += 256) { float t = w[i]; s += t * t; }
  sh[threadIdx.x] = s;
  __syncthreads();
  for (int off = 128; off > 0; off >>= 1) {
    if ((int)threadIdx.x < off) sh[threadIdx.x] += sh[threadIdx.x + off];
    __syncthreads();
  }
  if (threadIdx.x == 0) out[0] = sqrtf(sh[0]);
}

__global__ void score_k(const float* __restrict__ x, const float* __restrict__ w,
                        const float* __restrict__ nrm, float* __restrict__ s,
                        int n, int nvalid, int cout)
{
  int i = blockIdx.x * blockDim.x + threadIdx.x;
  if (i >= n) return;
  if (i >= nvalid) { s[i] = -3.3e38f; return; }
  float d = 0.0f;
  for (int c = 0; c < cout; ++c) d += x[(size_t)i * cout + c] * w[c];
  s[i] = tanhf(d / nrm[0]);
}

// rank[i] = #{ j : s[j] > s[i]  or (s[j]==s[i] and j<i) }  (sorted position)
__global__ void rank_k(const float* __restrict__ s, int* __restrict__ rank, int n) {
  int i = blockIdx.x * blockDim.x + threadIdx.x;
  if (i >= n) return;
  float si = s[i];
  int cnt = 0;
  for (int j = 0; j < n; ++j) {
    float sj = s[j];
    cnt += (sj > si) || (sj == si && j < i);
  }
  rank[i] = cnt;
}

__global__ void pool_scatter(const float* __restrict__ x, const float* __restrict__ s,
                             const int* __restrict__ rank, const float* __restrict__ pos,
                             float* __restrict__ xn, float* __restrict__ posn,
                             int* __restrict__ o2n, int n, int cout, int kkeep)
{
  int i = blockIdx.x * blockDim.x + threadIdx.x;
  if (i >= n) return;
  int r = rank[i];
  if (r < kkeep) {
    float sc = s[i];
    for (int c = 0; c < cout; ++c)
      xn[(size_t)r * cout + c] = x[(size_t)i * cout + c] * sc;
    posn[2 * r]     = pos[2 * i];
    posn[2 * r + 1] = pos[2 * i + 1];
    o2n[i] = r;
  } else {
    o2n[i] = kkeep;
  }
  if (i == 0) {                                  // dummy node
    for (int c = 0; c < cout; ++c) xn[(size_t)kkeep * cout + c] = 0.0f;
    posn[2 * kkeep] = 1e9f;
    posn[2 * kkeep + 1] = 1e9f;
  }
}

__global__ void edge_remap(const int* __restrict__ ei, const float* __restrict__ ea,
                           const int* __restrict__ o2n,
                           int* __restrict__ ein, float* __restrict__ ean,
                           int E, int kkeep)
{
  int e = blockIdx.x * blockDim.x + threadIdx.x;
  if (e >= E) return;
  int s = o2n[ei[e]];
  int d = o2n[ei[E + e]];
  bool bad = (s == kkeep) || (d == kkeep);
  ein[e]     = bad ? kkeep : s;
  ein[E + e] = bad ? kkeep : d;
  ean[2 * e]     = bad ? 0.0f : ea[2 * e];
  ean[2 * e + 1] = bad ? 0.0f : ea[2 * e + 1];
}

// brute-force 3-NN inverse-distance^2 interpolation; one thread per target.
__global__ void knn3_interp(const float* __restrict__ posx, int nx,
                            const float* __restrict__ posy, int ny,
                            const float* __restrict__ xsrc,
                            float* __restrict__ out, int cout)
{
  int y = blockIdx.x * blockDim.x + threadIdx.x;
  if (y >= ny) return;
  float py0 = posy[2 * y], py1 = posy[2 * y + 1];
  float d0 = 3.3e38f, d1 = 3.3e38f, d2 = 3.3e38f;
  int   i0 = 0, i1 = 0, i2 = 0;
  for (int j = 0; j < nx; ++j) {
    if (j + 32 < nx) __builtin_prefetch(&posx[2 * (j + 32)], 0, 1);
    float dx = py0 - posx[2 * j];
    float dy = py1 - posx[2 * j + 1];
    float d = dx * dx + dy * dy;
    if (d < d0)      { d2 = d1; i2 = i1; d1 = d0; i1 = i0; d0 = d; i0 = j; }
    else if (d < d1) { d2 = d1; i2 = i1; d1 = d;  i1 = j; }
    else if (d < d2) { d2 = d;  i2 = j; }
  }
  float w0 = 1.0f / fmaxf(d0, 1e-16f);
  float w1 = 1.0f / fmaxf(d1, 1e-16f);
  float w2 = 1.0f / fmaxf(d2, 1e-16f);
  float ws = w0 + w1 + w2;
  for (int c = 0; c < cout; ++c)
    out[(size_t)y * cout + c] =
        (w0 * xsrc[(size_t)i0 * cout + c] +
         w1 * xsrc[(size_t)i1 * cout + c] +
         w2 * xsrc[(size_t)i2 * cout + c]) / ws;
}

// ---------------------------- host helpers ---------------------------------

static inline void launch_gemm(hipStream_t st, const float* A, int lda,
                               const float* B, int ldb, float* C, int ldc,
                               int M, int N, int K)
{
  int tiles = ((M + 15) / 16) * ((N + 15) / 16);
  int blocks = (tiles + 7) / 8;                  // 8 waves (16x16 tiles) per block
  gemm_wmma_f32<<<blocks, 256, 0, st>>>(A, lda, B, ldb, C, ldc, M, N, K);
}

static inline void launch_fill(hipStream_t st, float* p, float v, int n) {
  fill_f32<<<(n + 255) / 256, 256, 0, st>>>(p, v, n);
}

// full SplineConv (mean aggr + root + bias [+ ELU] [+ accumulate])
static void run_spline(hipStream_t st,
                       const float* xin, int n, int cin, int cout, int K,
                       const int* ei, const float* ea, int E,
                       const float* w, const float* root, const float* bias,
                       float* xw, float* agg, float* deg, float* rootb,
                       float* xout, int do_elu, int do_add)
{
  int ktot = K * K;
  for (int k = 0; k < ktot; ++k)
    launch_gemm(st, xin, cin, w + (size_t)k * cin * cout, cout,
                xw + (size_t)k * cout, ktot * cout, n, cout, cin);
  launch_gemm(st, xin, cin, root, cout, rootb, cout, n, cout, cin);
  launch_fill(st, agg, 0.0f, n * cout);
  launch_fill(st, deg, 0.0f, n);
  long long total = (long long)E * cout;
  edge_msg<<<(int)((total + 255) / 256), 256, 0, st>>>(xw, ei, ea, agg, deg,
                                                       E, cout, K, ktot);
  finalize_conv<<<(n * cout + 255) / 256, 256, 0, st>>>(xout, agg, deg, rootb,
                                                        bias, n, cout, do_elu, do_add);
}

static void run_pool(hipStream_t st, const float* x, int n, int cout,
                     int nvalid, int kkeep, const float* wpool,
                     const int* ei, const float* ea, int E, const float* pos,
                     float* xn, float* posn, int* ein, float* ean,
                     float* score, int* rank, int* o2n, float* nrm)
{
  wnorm_k<<<1, 256, 0, st>>>(wpool, cout, nrm);
  score_k<<<(n + 255) / 256, 256, 0, st>>>(x, wpool, nrm, score, n, nvalid, cout);
  rank_k<<<(n + 255) / 256, 256, 0, st>>>(score, rank, n);
  pool_scatter<<<(n + 255) / 256, 256, 0, st>>>(x, score, rank, pos, xn, posn,
                                                o2n, n, cout, kkeep);
  edge_remap<<<(E + 255) / 256, 256, 0, st>>>(ei, ea, o2n, ein, ean, E, kkeep);
}

// Flattened param indices: inputs 0-4 are x, edge_index, edge_attr, pos, batch;
// params leaves follow in alphabetical tree order (jax tree_flatten).
enum {
  P_CONVOUT_B = 5,  P_CONVOUT_R, P_CONVOUT_W,
  P_D0A_B, P_D0A_R, P_D0A_W,
  P_D0B_B, P_D0B_R, P_D0B_W,
  P_D0BN_B, P_D0BN_G, P_D0BN_M, P_D0BN_V,
  P_D0POOL,
  P_D1A_B, P_D1A_R, P_D1A_W,
  P_D1B_B, P_D1B_R, P_D1B_W,
  P_D1BN_B, P_D1BN_G, P_D1BN_M, P_D1BN_V,
  P_D1POOL,
  P_D2A_B, P_D2A_R, P_D2A_W,
  P_D2B_B, P_D2B_R, P_D2B_W,
  P_D2BN_B, P_D2BN_G, P_D2BN_M, P_D2BN_V,
  P_D2POOL,
  P_D3A_B, P_D3A_R, P_D3A_W,
  P_D3B_B, P_D3B_R, P_D3B_W,
  P_D3BN_B, P_D3BN_G, P_D3BN_M, P_D3BN_V,
  P_D3POOL,
  P_SCOREFS_B, P_SCOREFS_R, P_SCOREFS_W,
  P_SCOREP2_B, P_SCOREP2_R, P_SCOREP2_W,
  P_SCOREP3_B, P_SCOREP3_R, P_SCOREP3_W,
  P_UP1_B, P_UP1_R, P_UP1_W
};

extern "C" void kernel_launch(void* const* d_in, const int* in_sizes, int n_in,
                              void* d_out, int out_size, void* d_ws, size_t ws_size,
                              hipStream_t stream)
{
  (void)in_sizes; (void)n_in; (void)out_size; (void)ws_size;
  hipStream_t st = stream;

  const float* X0   = (const float*)d_in[0];
  const int*   EI0  = (const int*)d_in[1];
  const float* EA0  = (const float*)d_in[2];
  const float* POS0 = (const float*)d_in[3];
  auto F = [&](int i) { return (const float*)d_in[i]; };

  const int E = EDG;
  // node counts per level (incl. dummy) and keep counts
  // L0:16384 -> L1:8193 -> L2:4097 -> L3:2049 -> L4:1025

  // --- workspace bump allocator (deterministic layout each call) ---
  char* wp = (char*)d_ws;
  auto af = [&](size_t n) { float* p = (float*)wp; wp += n * sizeof(float); return p; };
  auto ai = [&](size_t n) { int*   p = (int*)wp;   wp += n * sizeof(int);   return p; };

  float* XW    = af(13200000);   // max xw: 16384*25*32
  float* AGG   = af(600000);
  float* ROOTB = af(600000);
  float* DEG   = af(16384);
  float* XA    = af(600000);
  float* XB    = af(600000);
  float* XS2   = af(270000);     // saved pool2 features 4097*64
  float* XS3   = af(270000);     // saved pool3 features 2049*128
  float* UP    = af(530000);     // decoder features (<=16384*32)
  float* UT    = af(530000);
  float* SCORE = af(16384);
  float* NRM   = af(16);
  int*   RANK  = ai(16384);
  int*   O2N   = ai(16384);
  int*   EI1 = ai(2 * E); int* EI2 = ai(2 * E); int* EI3 = ai(2 * E); int* EI4 = ai(2 * E);
  float* EA1 = af(2 * E); float* EA2 = af(2 * E); float* EA3 = af(2 * E); float* EA4 = af(2 * E);
  float* POS1 = af(2 * 8193); float* POS2 = af(2 * 4097);
  float* POS3 = af(2 * 2049); float* POS4 = af(2 * 1025);

  // ---------------- encoder ----------------
  // L0: d0a (1->32, K=5), d0b (32->32, K=3), BN, pool -> 8193
  run_spline(st, X0, 16384, 1, 32, 5, EI0, EA0, E,
             F(P_D0A_W), F(P_D0A_R), F(P_D0A_B), XW, AGG, DEG, ROOTB, XA, 1, 0);
  run_spline(st, XA, 16384, 32, 32, 3, EI0, EA0, E,
             F(P_D0B_W), F(P_D0B_R), F(P_D0B_B), XW, AGG, DEG, ROOTB, XB, 1, 0);
  bn_apply<<<(16384 * 32 + 255) / 256, 256, 0, st>>>(
      XB, F(P_D0BN_G), F(P_D0BN_B), F(P_D0BN_M), F(P_D0BN_V), 16384, 32);
  run_pool(st, XB, 16384, 32, 16384, 8192, F(P_D0POOL), EI0, EA0, E, POS0,
           XA, POS1, EI1, EA1, SCORE, RANK, O2N, NRM);

  // L1: 8193 nodes, d1a (32->64), d1b (64->64), BN, pool -> 4097
  run_spline(st, XA, 8193, 32, 64, 3, EI1, EA1, E,
             F(P_D1A_W), F(P_D1A_R), F(P_D1A_B), XW, AGG, DEG, ROOTB, XB, 1, 0);
  run_spline(st, XB, 8193, 64, 64, 3, EI1, EA1, E,
             F(P_D1B_W), F(P_D1B_R), F(P_D1B_B), XW, AGG, DEG, ROOTB, XA, 1, 0);
  bn_apply<<<(8193 * 64 + 255) / 256, 256, 0, st>>>(
      XA, F(P_D1BN_G), F(P_D1BN_B), F(P_D1BN_M), F(P_D1BN_V), 8193, 64);
  run_pool(st, XA, 8193, 64, 8192, 4096, F(P_D1POOL), EI1, EA1, E, POS1,
           XB, POS2, EI2, EA2, SCORE, RANK, O2N, NRM);
  hipMemcpyAsync(XS2, XB, (size_t)4097 * 64 * sizeof(float),
                 hipMemcpyDeviceToDevice, st);            // pool_saved[2]

  // L2: 4097 nodes, d2a (64->128), d2b (128->128), BN, pool -> 2049
  run_spline(st, XB, 4097, 64, 128, 3, EI2, EA2, E,
             F(P_D2A_W), F(P_D2A_R), F(P_D2A_B), XW, AGG, DEG, ROOTB, XA, 1, 0);
  run_spline(st, XA, 4097, 128, 128, 3, EI2, EA2, E,
             F(P_D2B_W), F(P_D2B_R), F(P_D2B_B), XW, AGG, DEG, ROOTB, XB, 1, 0);
  bn_apply<<<(4097 * 128 + 255) / 256, 256, 0, st>>>(
      XB, F(P_D2BN_G), F(P_D2BN_B), F(P_D2BN_M), F(P_D2BN_V), 4097, 128);
  run_pool(st, XB, 4097, 128, 4096, 2048, F(P_D2POOL), EI2, EA2, E, POS2,
           XA, POS3, EI3, EA3, SCORE, RANK, O2N, NRM);
  hipMemcpyAsync(XS3, XA, (size_t)2049 * 128 * sizeof(float),
                 hipMemcpyDeviceToDevice, st);            // pool_saved[3]

  // L3: 2049 nodes, d3a (128->256, K=1), d3b (256->256, K=1), BN, pool -> 1025
  run_spline(st, XA, 2049, 128, 256, 1, EI3, EA3, E,
             F(P_D3A_W), F(P_D3A_R), F(P_D3A_B), XW, AGG, DEG, ROOTB, XB, 1, 0);
  run_spline(st, XB, 2049, 256, 256, 1, EI3, EA3, E,
             F(P_D3B_W), F(P_D3B_R), F(P_D3B_B), XW, AGG, DEG, ROOTB, XA, 1, 0);
  bn_apply<<<(2049 * 256 + 255) / 256, 256, 0, st>>>(
      XA, F(P_D3BN_G), F(P_D3BN_B), F(P_D3BN_M), F(P_D3BN_V), 2049, 256);
  run_pool(st, XA, 2049, 256, 2048, 1024, F(P_D3POOL), EI3, EA3, E, POS3,
           XB, POS4, EI4, EA4, SCORE, RANK, O2N, NRM);

  // ---------------- decoder ----------------
  // score_fs (256->32) + up1_conv (32->32) on L4 graph
  run_spline(st, XB, 1025, 256, 32, 3, EI4, EA4, E,
             F(P_SCOREFS_W), F(P_SCOREFS_R), F(P_SCOREFS_B),
             XW, AGG, DEG, ROOTB, XA, 1, 0);
  run_spline(st, XA, 1025, 32, 32, 3, EI4, EA4, E,
             F(P_UP1_W), F(P_UP1_R), F(P_UP1_B), XW, AGG, DEG, ROOTB, XB, 1, 0);

  // interp L4 -> L3; add elu(score_pool3(saved L3 features))
  knn3_interp<<<(2049 + 255) / 256, 256, 0, st>>>(POS4, 1025, POS3, 2049, XB, UP, 32);
  run_spline(st, XS3, 2049, 128, 32, 3, EI3, EA3, E,
             F(P_SCOREP3_W), F(P_SCOREP3_R), F(P_SCOREP3_B),
             XW, AGG, DEG, ROOTB, UP, 1, 1);

  // interp L3 -> L2; add elu(score_pool2(saved L2 features))
  knn3_interp<<<(4097 + 255) / 256, 256, 0, st>>>(POS3, 2049, POS2, 4097, UP, UT, 32);
  run_spline(st, XS2, 4097, 64, 32, 3, EI2, EA2, E,
             F(P_SCOREP2_W), F(P_SCOREP2_R), F(P_SCOREP2_B),
             XW, AGG, DEG, ROOTB, UT, 1, 1);

  // interp L2 -> L1; up1_conv (shared weights) on L1 graph
  knn3_interp<<<(8193 + 255) / 256, 256, 0, st>>>(POS2, 4097, POS1, 8193, UT, UP, 32);
  run_spline(st, UP, 8193, 32, 32, 3, EI1, EA1, E,
             F(P_UP1_W), F(P_UP1_R), F(P_UP1_B), XW, AGG, DEG, ROOTB, UT, 1, 0);

  // interp L1 -> L0; final convout (32->1, K=5), no ELU, into d_out
  knn3_interp<<<(16384 + 255) / 256, 256, 0, st>>>(POS1, 8193, POS0, 16384, UT, UP, 32);
  run_spline(st, UP, 16384, 32, 1, 5, EI0, EA0, E,
             F(P_CONVOUT_W), F(P_CONVOUT_R), F(P_CONVOUT_B),
             XW, AGG, DEG, ROOTB, (float*)d_out, 0, 0);
}